// GAT_LSTM_2680059593213
// MI455X (gfx1250) — compile-verified
//
#include <hip/hip_runtime.h>
#include <math.h>

typedef __attribute__((ext_vector_type(2))) float v2f;
typedef __attribute__((ext_vector_type(8))) float v8f;

#define NNODE 8192
#define BATCH 4
#define SDIM  12
#define NEDGE 131072
#define NROW  32768           // BATCH * NNODE
#define LDIM  5

__device__ __forceinline__ float sigmoidf_(float x) { return 1.0f / (1.0f + __expf(-x)); }

// ---------------------------------------------------------------------------
// xp[r*12+t] = x[b][t][n] with r = b*8192+n   (x is (B,S,N))
__global__ void transpose_x_kernel(const float* __restrict__ x, float* __restrict__ xp) {
  int idx = blockIdx.x * blockDim.x + threadIdx.x;
  if (idx >= NROW * SDIM) return;
  int r = idx / SDIM, t = idx - r * SDIM;
  int b = r >> 13, n = r & (NNODE - 1);
  xp[idx] = x[(b * SDIM + t) * NNODE + n];
}

// p = W @ a[:12], q = W @ a[12:]
__global__ void compute_pq_kernel(const float* __restrict__ W, const float* __restrict__ a,
                                  float* __restrict__ pq) {
  int t = threadIdx.x;
  if (t < SDIM) {
    float p = 0.f, q = 0.f;
    for (int c = 0; c < SDIM; ++c) {
      p += W[t * SDIM + c] * a[c];
      q += W[t * SDIM + c] * a[SDIM + c];
    }
    pq[t] = p;
    pq[SDIM + t] = q;
  }
}

// per-node u,v using batch-0 rows of fin
__global__ void node_uv_kernel(const float* __restrict__ fin, const float* __restrict__ pq,
                               float* __restrict__ u, float* __restrict__ v) {
  int n = blockIdx.x * blockDim.x + threadIdx.x;
  if (n >= NNODE) return;
  float su = 0.f, sv = 0.f;
  #pragma unroll
  for (int r = 0; r < SDIM; ++r) {
    float f = fin[n * SDIM + r];
    su += f * pq[r];
    sv += f * pq[SDIM + r];
  }
  u[n] = su;
  v[n] = sv;
}

// per-edge: w = exp(leaky(u+v)*lw + lb); colsum[j] += w
__global__ void edge_exp_kernel(const int* __restrict__ e, const float* __restrict__ u,
                                const float* __restrict__ v, const float* __restrict__ lw,
                                const float* __restrict__ lb, float* __restrict__ w,
                                float* __restrict__ colsum) {
  int k = blockIdx.x * blockDim.x + threadIdx.x;
  if (k >= NEDGE) return;
  int i = e[k], j = e[NEDGE + k];
  float val = u[i] + v[j];
  val = (val > 0.f) ? val : 0.2f * val;
  val = val * lw[0] + lb[0];
  float wk = __expf(val);
  w[k] = wk;
  atomicAdd(&colsum[j], wk);
}

// sparse att @ features: z[b,i,:] += (w/colsum[j]) * fin[b,j,:]
__global__ void spmm_kernel(const int* __restrict__ e, const float* __restrict__ w,
                            const float* __restrict__ colsum, const float* __restrict__ fin,
                            float* __restrict__ z) {
  int k = blockIdx.x * blockDim.x + threadIdx.x;
  if (k >= NEDGE) return;
  int i = e[k], j = e[NEDGE + k];
  float att = w[k] / colsum[j];
  #pragma unroll
  for (int b = 0; b < BATCH; ++b) {
    const float* src = fin + (b * NNODE + j) * SDIM;
    float* dst = z + (b * NNODE + i) * SDIM;
    #pragma unroll
    for (int t = 0; t < SDIM; ++t) atomicAdd(&dst[t], att * src[t]);
  }
}

// empty attention columns -> uniform 1/N softmax row contribution
__global__ void empty_cols_kernel(const float* __restrict__ colsum, const float* __restrict__ fin,
                                  float* __restrict__ uni) {
  int j = blockIdx.x * blockDim.x + threadIdx.x;
  if (j >= NNODE) return;
  if (colsum[j] == 0.0f) {
    const float inv = 1.0f / (float)NNODE;
    for (int b = 0; b < BATCH; ++b)
      for (int t = 0; t < SDIM; ++t)
        atomicAdd(&uni[b * SDIM + t], inv * fin[(b * NNODE + j) * SDIM + t]);
  }
}

// GCN: out = leaky_relu((z+uni) @ gcnW^T + gcnb, 0.01) via V_WMMA_F32_16X16X4_F32
// one wave32 per 16-row tile; K=12 = 3 chunks of 4; N=12 padded to 16.
// B loads are unconditional (clamped address) + lane select: no exec divergence
// between the chained WMMAs.
__global__ void gcn_wmma_kernel(const float* __restrict__ z, const float* __restrict__ uni,
                                const float* __restrict__ gcnW, const float* __restrict__ gcnb,
                                float* __restrict__ out) {
  int wave = (blockIdx.x * blockDim.x + threadIdx.x) >> 5;  // 2048 tiles total
  int lane = threadIdx.x & 31;
  int m = lane & 15;        // A row within tile / B,D column
  int khalf = lane >> 4;    // K split across lane halves
  int row = wave * 16 + m;
  int b = row >> 13;        // batch of this tile (tiles never straddle batches)
  int mc = (m < SDIM) ? m : (SDIM - 1);   // clamped column for safe unconditional loads
  float colmask = (m < SDIM) ? 1.0f : 0.0f;
  v8f acc = {};
  #pragma unroll
  for (int kc = 0; kc < 3; ++kc) {
    int k0 = kc * 4 + 2 * khalf;          // even -> float2 loads are 8B aligned
    // A 16x4 f32 layout: VGPR0 = K{0|2}, VGPR1 = K{1|3} per lane-half
    v2f av = *(const v2f*)(z + row * SDIM + k0);
    v2f uv = *(const v2f*)(uni + b * SDIM + k0);
    av.x += uv.x;
    av.y += uv.y;
    // B 4x16 f32: B[k][n] = gcnW[n][k], n = lane&15; pad n>=12 with 0 via select
    v2f bv = *(const v2f*)(gcnW + mc * SDIM + k0);
    bv.x *= colmask;
    bv.y *= colmask;
    acc = __builtin_amdgcn_wmma_f32_16x16x4_f32(false, av, false, bv,
                                                (short)0, acc, false, false);
  }
  // D layout: lane holds column n = lane&15, rows m = vg + 8*(lane>>4)
  int n = lane & 15;
  float bias = gcnb[mc];                  // clamped, unconditional
  if (n < SDIM) {
    #pragma unroll
    for (int vg = 0; vg < 8; ++vg) {
      int rr = wave * 16 + vg + 8 * khalf;
      float val = acc[vg] + bias;
      out[rr * SDIM + n] = (val > 0.f) ? val : 0.01f * val;
    }
  }
}

// fused 2-layer LSTM, hidden=5, 12 steps; one thread per row
__global__ void lstm_kernel(const float* __restrict__ x, const float* __restrict__ c1,
                            const float* __restrict__ c2, const float* __restrict__ c4,
                            const float* __restrict__ c5, const float* __restrict__ Wih0,
                            const float* __restrict__ Whh0, const float* __restrict__ bih0,
                            const float* __restrict__ bhh0, const float* __restrict__ Wih1,
                            const float* __restrict__ Whh1, const float* __restrict__ bih1,
                            const float* __restrict__ bhh1, float* __restrict__ lstm_out) {
  __shared__ float sWi0[100], sWh0[100], sWi1[100], sWh1[100], sB0[20], sB1[20];
  for (int i = threadIdx.x; i < 100; i += blockDim.x) {
    sWi0[i] = Wih0[i]; sWh0[i] = Whh0[i]; sWi1[i] = Wih1[i]; sWh1[i] = Whh1[i];
  }
  for (int i = threadIdx.x; i < 20; i += blockDim.x) {
    sB0[i] = bih0[i] + bhh0[i];
    sB1[i] = bih1[i] + bhh1[i];
  }
  __syncthreads();
  int r = blockIdx.x * blockDim.x + threadIdx.x;
  if (r >= NROW) return;
  int b = r >> 13, n = r & (NNODE - 1);
  float h0[5] = {}, s0[5] = {}, h1[5] = {}, s1[5] = {};
  for (int t = 0; t < SDIM; ++t) {
    float xt[5];
    xt[0] = x[(b * SDIM + t) * NNODE + n];
    xt[1] = c1[r * SDIM + t];
    xt[2] = c2[r * SDIM + t];
    xt[3] = c4[r * SDIM + t];
    xt[4] = c5[r * SDIM + t];
    float zi[5], zf[5], zg[5], zo[5];
    // layer 0 gates (i,f,g,o)
    #pragma unroll
    for (int l = 0; l < 5; ++l) {
      float a0 = sB0[l], a1 = sB0[5 + l], a2 = sB0[10 + l], a3 = sB0[15 + l];
      #pragma unroll
      for (int mm = 0; mm < 5; ++mm) {
        a0 += xt[mm] * sWi0[l * 5 + mm]        + h0[mm] * sWh0[l * 5 + mm];
        a1 += xt[mm] * sWi0[(5 + l) * 5 + mm]  + h0[mm] * sWh0[(5 + l) * 5 + mm];
        a2 += xt[mm] * sWi0[(10 + l) * 5 + mm] + h0[mm] * sWh0[(10 + l) * 5 + mm];
        a3 += xt[mm] * sWi0[(15 + l) * 5 + mm] + h0[mm] * sWh0[(15 + l) * 5 + mm];
      }
      zi[l] = a0; zf[l] = a1; zg[l] = a2; zo[l] = a3;
    }
    #pragma unroll
    for (int l = 0; l < 5; ++l) {
      s0[l] = sigmoidf_(zf[l]) * s0[l] + sigmoidf_(zi[l]) * tanhf(zg[l]);
      h0[l] = sigmoidf_(zo[l]) * tanhf(s0[l]);
    }
    // layer 1 gates (input = new h0)
    #pragma unroll
    for (int l = 0; l < 5; ++l) {
      float a0 = sB1[l], a1 = sB1[5 + l], a2 = sB1[10 + l], a3 = sB1[15 + l];
      #pragma unroll
      for (int mm = 0; mm < 5; ++mm) {
        a0 += h0[mm] * sWi1[l * 5 + mm]        + h1[mm] * sWh1[l * 5 + mm];
        a1 += h0[mm] * sWi1[(5 + l) * 5 + mm]  + h1[mm] * sWh1[(5 + l) * 5 + mm];
        a2 += h0[mm] * sWi1[(10 + l) * 5 + mm] + h1[mm] * sWh1[(10 + l) * 5 + mm];
        a3 += h0[mm] * sWi1[(15 + l) * 5 + mm] + h1[mm] * sWh1[(15 + l) * 5 + mm];
      }
      zi[l] = a0; zf[l] = a1; zg[l] = a2; zo[l] = a3;
    }
    #pragma unroll
    for (int l = 0; l < 5; ++l) {
      s1[l] = sigmoidf_(zf[l]) * s1[l] + sigmoidf_(zi[l]) * tanhf(zg[l]);
      h1[l] = sigmoidf_(zo[l]) * tanhf(s1[l]);
      lstm_out[r * (SDIM * LDIM) + t * LDIM + l] = h1[l];
    }
  }
}

// fused TPA head; one thread per row. out[0:32768]=y, out[32768:]=a (r,l)
__global__ void tpa_kernel(const float* __restrict__ lo, const float* __restrict__ fc1W,
                           const float* __restrict__ fc1b, const float* __restrict__ fc2W,
                           const float* __restrict__ fc2b, const float* __restrict__ fc3W,
                           const float* __restrict__ fc3b, float* __restrict__ out) {
  __shared__ float s1W[176], s1b[16], s2W[80], s2b[5], s3W[21], s3b[1];
  for (int i = threadIdx.x; i < 176; i += blockDim.x) s1W[i] = fc1W[i];
  for (int i = threadIdx.x; i < 80; i += blockDim.x) s2W[i] = fc2W[i];
  if (threadIdx.x < 16) s1b[threadIdx.x] = fc1b[threadIdx.x];
  if (threadIdx.x < 5) s2b[threadIdx.x] = fc2b[threadIdx.x];
  if (threadIdx.x < 21) s3W[threadIdx.x] = fc3W[threadIdx.x];
  if (threadIdx.x == 0) s3b[0] = fc3b[0];
  __syncthreads();
  int r = blockIdx.x * blockDim.x + threadIdx.x;
  if (r >= NROW) return;
  const float* row = lo + r * (SDIM * LDIM);
  float ht[5];
  #pragma unroll
  for (int m = 0; m < 5; ++m) ht[m] = row[11 * LDIM + m];
  float Hc[5][16];
  #pragma unroll
  for (int l = 0; l < 5; ++l)
    for (int k = 0; k < 16; ++k) {
      float s = s1b[k];
      #pragma unroll
      for (int t = 0; t < 11; ++t) s += row[t * LDIM + l] * s1W[k * 11 + t];
      Hc[l][k] = s;
    }
  float av[5];
  #pragma unroll
  for (int l = 0; l < 5; ++l) {
    float s = 0.f;
    #pragma unroll
    for (int m = 0; m < 5; ++m) {
      float hn = s2b[m];
      #pragma unroll
      for (int k = 0; k < 16; ++k) hn += Hc[l][k] * s2W[m * 16 + k];
      s += hn * ht[m];
    }
    av[l] = sigmoidf_(s);
  }
  float y = s3b[0];
  #pragma unroll
  for (int k = 0; k < 16; ++k) {
    float vt = 0.f;
    #pragma unroll
    for (int l = 0; l < 5; ++l) vt += av[l] * Hc[l][k];
    y += s3W[k] * vt;
  }
  #pragma unroll
  for (int m = 0; m < 5; ++m) y += s3W[16 + m] * ht[m];
  out[r] = y;
  #pragma unroll
  for (int l = 0; l < 5; ++l) out[NROW + r * LDIM + l] = av[l];
}

// ---------------------------------------------------------------------------
extern "C" void kernel_launch(void* const* d_in, const int* in_sizes, int n_in,
                              void* d_out, int out_size, void* d_ws, size_t ws_size,
                              hipStream_t stream) {
  const float* x    = (const float*)d_in[0];
  const int*   e1   = (const int*)d_in[1];
  const int*   e2   = (const int*)d_in[2];
  const float* g1W  = (const float*)d_in[3];
  const float* g1a  = (const float*)d_in[4];
  const float* g1lw = (const float*)d_in[5];
  const float* g1lb = (const float*)d_in[6];
  const float* g2W  = (const float*)d_in[7];
  const float* g2a  = (const float*)d_in[8];
  const float* g2lw = (const float*)d_in[9];
  const float* g2lb = (const float*)d_in[10];
  const float* gcnW = (const float*)d_in[11];
  const float* gcnb = (const float*)d_in[12];
  const float* Wih0 = (const float*)d_in[13];
  const float* Whh0 = (const float*)d_in[14];
  const float* bih0 = (const float*)d_in[15];
  const float* bhh0 = (const float*)d_in[16];
  const float* Wih1 = (const float*)d_in[17];
  const float* Whh1 = (const float*)d_in[18];
  const float* bih1 = (const float*)d_in[19];
  const float* bhh1 = (const float*)d_in[20];
  const float* fc1W = (const float*)d_in[21];
  const float* fc1b = (const float*)d_in[22];
  const float* fc2W = (const float*)d_in[23];
  const float* fc2b = (const float*)d_in[24];
  const float* fc3W = (const float*)d_in[25];
  const float* fc3b = (const float*)d_in[26];

  const size_t FEAT = (size_t)NROW * SDIM;  // 393216 floats
  float* ws  = (float*)d_ws;
  float* XP  = ws;
  float* C1  = XP + FEAT;
  float* C2  = C1 + FEAT;
  float* C4  = C2 + FEAT;
  float* C5  = C4 + FEAT;
  float* Z   = C5 + FEAT;
  float* LO  = Z + FEAT;                    // 32768*12*5 = 1966080 floats
  float* WB  = LO + (size_t)NROW * SDIM * LDIM;
  float* CS  = WB + NEDGE;
  float* U   = CS + NNODE;
  float* V   = U + NNODE;
  float* PQ  = V + NNODE;
  float* UNI = PQ + 32;

  transpose_x_kernel<<<(NROW * SDIM + 255) / 256, 256, 0, stream>>>(x, XP);

  auto stage = [&](const float* fin, float* fout, const int* e, const float* W,
                   const float* a, const float* lw, const float* lb) {
    hipMemsetAsync(Z, 0, FEAT * sizeof(float), stream);
    hipMemsetAsync(CS, 0, NNODE * sizeof(float), stream);
    hipMemsetAsync(UNI, 0, 64 * sizeof(float), stream);
    compute_pq_kernel<<<1, 32, 0, stream>>>(W, a, PQ);
    node_uv_kernel<<<NNODE / 256, 256, 0, stream>>>(fin, PQ, U, V);
    edge_exp_kernel<<<NEDGE / 256, 256, 0, stream>>>(e, U, V, lw, lb, WB, CS);
    spmm_kernel<<<NEDGE / 256, 256, 0, stream>>>(e, WB, CS, fin, Z);
    empty_cols_kernel<<<NNODE / 256, 256, 0, stream>>>(CS, fin, UNI);
    gcn_wmma_kernel<<<256, 256, 0, stream>>>(Z, UNI, gcnW, gcnb, fout);  // 2048 waves
  };

  stage(XP, C1, e1, g1W, g1a, g1lw, g1lb);
  stage(C1, C2, e1, g1W, g1a, g1lw, g1lb);
  stage(XP, C4, e2, g2W, g2a, g2lw, g2lb);
  stage(C4, C5, e2, g2W, g2a, g2lw, g2lb);

  lstm_kernel<<<NROW / 256, 256, 0, stream>>>(x, C1, C2, C4, C5, Wih0, Whh0, bih0, bhh0,
                                              Wih1, Whh1, bih1, bhh1, LO);
  tpa_kernel<<<NROW / 256, 256, 0, stream>>>(LO, fc1W, fc1b, fc2W, fc2b, fc3W, fc3b,
                                             (float*)d_out);
}